// VectorQuantize_2619930051595
// MI455X (gfx1250) — compile-verified
//
#include <hip/hip_runtime.h>

typedef __attribute__((ext_vector_type(2))) float v2f;
typedef __attribute__((ext_vector_type(8))) float v8f;

#define NUM_EMB   1024
#define DIM       64
#define NVEC      131072          // 32*64*64
#define NTILES    (NVEC / 16)     // 8192
#define HW        4096            // 64*64
#define IMG       262144          // 64*64*64 floats per batch image of z_e
#define ZQ_SIZE   8388608         // 32*64*64*64
#define LOSS_OFF  ZQ_SIZE
#define IDX_OFF   (ZQ_SIZE + 1)
#define NKT       (NUM_EMB / 16)  // 64 code tiles

// ---------------------------------------------------------------------------
// Kernel 0: per-code squared norms  e2[k] = sum_d embed[k][d]^2   (1024 codes)
// ---------------------------------------------------------------------------
__global__ void vq_e2_kernel(const float* __restrict__ embed, float* __restrict__ e2) {
    int k = blockIdx.x * blockDim.x + threadIdx.x;
    if (k < NUM_EMB) {
        const float4* p = (const float4*)(embed + (long)k * DIM);
        float s = 0.f;
#pragma unroll
        for (int i = 0; i < DIM / 4; ++i) {
            float4 v = p[i];
            s += v.x * v.x + v.y * v.y + v.z * v.z + v.w * v.w;
        }
        e2[k] = s;
    }
}

// ---------------------------------------------------------------------------
// Kernel 1: main VQ. One wave32 per 16-vector tile; 8 waves per block share a
// double-buffered LDS 16-code x 64-dim B tile. Dot products via
// V_WMMA_F32_16X16X4_F32, two independent accumulator chains.
// ---------------------------------------------------------------------------
__launch_bounds__(256)
__global__ void vq_main_kernel(const float* __restrict__ z_e,
                               const float* __restrict__ embed,
                               const float* __restrict__ e2,
                               float* __restrict__ out,
                               float* __restrict__ partials) {
    __shared__ float bbuf[2][16 * 66]; // 16 code rows x 64 dims, +2 pad vs bank conflicts
    __shared__ int   idxbuf[8][16];    // winning code per (wave, row)

    const int tid  = threadIdx.x;
    const int lane = tid & 31;
    const int wave = tid >> 5;
    const int half = lane >> 4;       // selects K pair {2h, 2h+1} of each 4-chunk
    const int mrow = lane & 15;       // A-matrix row / B-matrix column

    const int  tile  = blockIdx.x * 8 + wave;
    const int  n0    = tile * 16;                 // first flat vector of tile
    const int  bimg  = n0 >> 12;                  // n0 / 4096
    const int  r0    = n0 & (HW - 1);             // h*W + w of first row
    const long zbase = (long)bimg * IMG + r0;

    // cooperative-stage addressing (thread -> 4 floats of the 16x64 tile)
    const int srow = (tid * 4) >> 6;
    const int scol = (tid * 4) & 63;

    // ---- Prologue: stage code tile 0 into buffer 0 ----
    {
        float4 v = *(const float4*)(embed + tid * 4);
        float* dst = &bbuf[0][srow * 66 + scol];
        dst[0] = v.x; dst[1] = v.y; dst[2] = v.z; dst[3] = v.w;
    }

    // ---- Preload A tile into registers (16x4 f32 WMMA A layout) ----
    // lane holds row m = mrow; chunk ch covers dims 4ch..4ch+3, this lane owns
    // dims {4ch + 2*half, 4ch + 2*half + 1}.
    v2f a[16];
#pragma unroll
    for (int ch = 0; ch < 16; ++ch) {
        int d0 = 4 * ch + 2 * half;
        a[ch].x = z_e[zbase + (long)d0 * HW + mrow];
        a[ch].y = z_e[zbase + (long)(d0 + 1) * HW + mrow];
    }

    float bestv[8];
    int   besti[8];
#pragma unroll
    for (int j = 0; j < 8; ++j) { bestv[j] = 3.402823e38f; besti[j] = 0; }

    // ---- Sweep the codebook in 16-code tiles (double-buffered LDS) ----
    for (int kt = 0; kt < NKT; ++kt) {
        const int cb = kt * 16;
        __syncthreads();   // buffer (kt&1) stores visible; prior reads retired

        // Issue next tile's global load early so it overlaps the WMMA block.
        float4 nxt;
        if (kt + 1 < NKT)
            nxt = *(const float4*)(embed + (long)(cb + 16) * DIM + tid * 4);

        const float* bb = bbuf[kt & 1];
        v8f c0 = {0.f, 0.f, 0.f, 0.f, 0.f, 0.f, 0.f, 0.f};
        v8f c1 = {0.f, 0.f, 0.f, 0.f, 0.f, 0.f, 0.f, 0.f};
#pragma unroll
        for (int ch = 0; ch < 16; ch += 2) {
            int d0 = 4 * ch + 2 * half;
            // B 4x16 layout mirrors A: lane holds column n = mrow, K pair {2h,2h+1}
            v2f b0 = *(const v2f*)&bb[mrow * 66 + d0];
            v2f b1 = *(const v2f*)&bb[mrow * 66 + d0 + 4];
            c0 = __builtin_amdgcn_wmma_f32_16x16x4_f32(
                     false, a[ch],     false, b0, (short)0, c0, false, false);
            c1 = __builtin_amdgcn_wmma_f32_16x16x4_f32(
                     false, a[ch + 1], false, b1, (short)0, c1, false, false);
        }

        // Park the prefetched tile in the alternate buffer (no barrier needed:
        // the other buffer is idle; next iteration's barrier publishes it).
        if (kt + 1 < NKT) {
            float* dst = &bbuf[(kt + 1) & 1][srow * 66 + scol];
            dst[0] = nxt.x; dst[1] = nxt.y; dst[2] = nxt.z; dst[3] = nxt.w;
        }

        // c0[j]+c1[j] = dot(row m = j + 8*half, code n = cb + mrow)
        const float e2v  = e2[cb + mrow];
        const int   code = cb + mrow;
#pragma unroll
        for (int j = 0; j < 8; ++j) {
            float s = e2v - 2.0f * (c0[j] + c1[j]);
            if (s < bestv[j]) { bestv[j] = s; besti[j] = code; }
        }
    }

    // ---- Argmin reduce across the 16 lanes sharing each row group ----
#pragma unroll
    for (int j = 0; j < 8; ++j) {
#pragma unroll
        for (int mask = 1; mask < 16; mask <<= 1) {
            float ov = __shfl_xor(bestv[j], mask, 32);
            int   oi = __shfl_xor(besti[j], mask, 32);
            if (ov < bestv[j] || (ov == bestv[j] && oi < besti[j])) {
                bestv[j] = ov; besti[j] = oi;
            }
        }
    }
    if (mrow == 0) {
#pragma unroll
        for (int j = 0; j < 8; ++j) {
            int m = half * 8 + j;
            idxbuf[wave][m] = besti[j];
            out[IDX_OFF + n0 + m] = (float)besti[j];   // indices as float
        }
    }
    __syncthreads();

    // ---- Phase 2: gather winners -> z_q_st, commitment-loss partial ----
    const int   myidx = idxbuf[wave][mrow];
    const long  ebase = (long)myidx * DIM;
    float lpart = 0.f;
#pragma unroll
    for (int ch = 0; ch < 16; ++ch) {
        int d0 = 4 * ch + 2 * half;
        v2f e = *(const v2f*)(embed + ebase + d0);
        out[zbase + (long)d0 * HW + mrow]       = e.x;  // z_q_st == z_q numerically
        out[zbase + (long)(d0 + 1) * HW + mrow] = e.y;
        float dx = a[ch].x - e.x;
        float dy = a[ch].y - e.y;
        lpart += dx * dx + dy * dy;
    }
#pragma unroll
    for (int mask = 1; mask < 32; mask <<= 1)
        lpart += __shfl_xor(lpart, mask, 32);
    if (lane == 0) partials[tile] = lpart;
}

// ---------------------------------------------------------------------------
// Kernel 2: deterministic fixed-order reduction of 8192 partials -> loss
// ---------------------------------------------------------------------------
__global__ void vq_loss_kernel(const float* __restrict__ partials, float* __restrict__ out) {
    __shared__ float red[256];
    int t = threadIdx.x;
    float s = 0.f;
    for (int i = 0; i < NTILES / 256; ++i) s += partials[t + i * 256];
    red[t] = s;
    __syncthreads();
    for (int off = 128; off > 0; off >>= 1) {
        if (t < off) red[t] += red[t + off];
        __syncthreads();
    }
    if (t == 0) out[LOSS_OFF] = red[0] / (float)ZQ_SIZE;
}

extern "C" void kernel_launch(void* const* d_in, const int* in_sizes, int n_in,
                              void* d_out, int out_size, void* d_ws, size_t ws_size,
                              hipStream_t stream) {
    const float* z_e   = (const float*)d_in[0];
    const float* embed = (const float*)d_in[1];
    float* out      = (float*)d_out;
    float* e2       = (float*)d_ws;       // 1024 floats
    float* partials = e2 + NUM_EMB;       // 8192 floats  (36 KB total ws use)

    vq_e2_kernel<<<(NUM_EMB + 255) / 256, 256, 0, stream>>>(embed, e2);
    vq_main_kernel<<<NTILES / 8, 256, 0, stream>>>(z_e, embed, e2, out, partials);
    vq_loss_kernel<<<1, 256, 0, stream>>>(partials, out);
}